// SimpleGCN_57896159150317
// MI455X (gfx1250) — compile-verified
//
#include <hip/hip_runtime.h>

typedef __attribute__((ext_vector_type(2))) float v2f;
typedef __attribute__((ext_vector_type(8))) float v8f;

#define D 128

// ---------------- degree / normalization ----------------
__global__ __launch_bounds__(256) void k_deg_init(float* __restrict__ deg, int n) {
    int i = blockIdx.x * 256 + threadIdx.x;
    if (i < n) deg[i] = 1.0f;               // self-loop
}

__global__ __launch_bounds__(256) void k_deg_count(const int* __restrict__ col,
                                                   float* __restrict__ deg, int e) {
    int i = blockIdx.x * 256 + threadIdx.x;
    if (i < e) atomicAdd(&deg[col[i]], 1.0f);
}

__global__ __launch_bounds__(256) void k_rsqrt(float* __restrict__ deg, int n) {
    int i = blockIdx.x * 256 + threadIdx.x;
    if (i < n) deg[i] = rsqrtf(deg[i]);
}

// ---------------- H = X @ W   (fp32 WMMA 16x16x4) ----------------
// 256 threads = 8 waves. Block covers 128 nodes; wave w owns rows [16w,16w+16)
// of the tile and all 8 column tiles (D=128). W is staged whole in LDS (64 KB).
__global__ __launch_bounds__(256) void k_gemm(const float* __restrict__ X,
                                              const float* __restrict__ W,
                                              float* __restrict__ H, int n) {
    __shared__ float lw[D * D];             // 64 KB
    const int tid = threadIdx.x;
    for (int i = tid * 4; i < D * D; i += 256 * 4)
        *(float4*)(lw + i) = *(const float4*)(W + i);
    __syncthreads();

    const int wave = tid >> 5;
    const int lane = tid & 31;
    const int lh = lane >> 4;               // lane half (0/1)
    const int lm = lane & 15;               // lane within half
    const int node0 = blockIdx.x * 128 + wave * 16;

    int arow = node0 + lm;                  // clamp reads; stores are masked
    if (arow >= n) arow = n - 1;
    const float* xr = X + (size_t)arow * D;

    v8f acc[8] = {};
    for (int k = 0; k < D; k += 4) {
        // A 16x4 fp32 frag: lane l holds (M=l&15, K = k + 2*(l>>4) + {0,1})
        v2f a = *(const v2f*)(xr + k + 2 * lh);
        const float* w0 = lw + (size_t)(k + 2 * lh) * D + lm;
#pragma unroll
        for (int nt = 0; nt < 8; ++nt) {
            // B 4x16 fp32 frag: VGPR v holds row K = k + v + 2*(l>>4), col = nt*16 + (l&15)
            v2f b;
            b.x = w0[nt * 16];
            b.y = w0[D + nt * 16];
            acc[nt] = __builtin_amdgcn_wmma_f32_16x16x4_f32(
                false, a, false, b, (short)0, acc[nt], false, false);
        }
    }

#pragma unroll
    for (int v = 0; v < 8; ++v) {
        int node = node0 + v + 8 * lh;      // D layout: M = v + 8*(l>>4)
        if (node < n) {
            float* hr = H + (size_t)node * D;
#pragma unroll
            for (int nt = 0; nt < 8; ++nt) hr[nt * 16 + lm] = acc[nt][v];
        }
    }
}

// ---------------- AGG[i] = dinv[i]^2 * H[i] + bias  (self-loop term) ----------------
__global__ __launch_bounds__(256) void k_self(const float* __restrict__ H,
                                              const float* __restrict__ dinv,
                                              const float* __restrict__ bias,
                                              float* __restrict__ AGG, int n) {
    int t = blockIdx.x * 256 + threadIdx.x;
    int i = t >> 5;                          // 32 lanes per node, 4 floats each
    if (i >= n) return;
    int f = (t & 31) * 4;
    float d2 = dinv[i] * dinv[i];
    float4 h = *(const float4*)(H + (size_t)i * D + f);
    float4 b = *(const float4*)(bias + f);
    float4 o = make_float4(h.x * d2 + b.x, h.y * d2 + b.y,
                           h.z * d2 + b.z, h.w * d2 + b.w);
    *(float4*)(AGG + (size_t)i * D + f) = o;
}

// ---------------- edge scatter: AGG[col] += H[row] * dinv[row]*dinv[col] ----------------
// One edge per wave; each lane handles a float4 feature block (L2-resident atomics).
__global__ __launch_bounds__(256) void k_scatter(const float* __restrict__ H,
                                                 const int* __restrict__ row,
                                                 const int* __restrict__ col,
                                                 const float* __restrict__ dinv,
                                                 float* __restrict__ AGG, int e) {
    long t = (long)blockIdx.x * 256 + threadIdx.x;
    long ei = t >> 5;
    if (ei >= e) return;
    int f = ((int)t & 31) * 4;
    int r = row[ei];
    int c = col[ei];
    float coef = dinv[r] * dinv[c];
    float4 h = *(const float4*)(H + (size_t)r * D + f);
    float* dst = AGG + (size_t)c * D + f;
    atomicAdd(dst + 0, h.x * coef);
    atomicAdd(dst + 1, h.y * coef);
    atomicAdd(dst + 2, h.z * coef);
    atomicAdd(dst + 3, h.w * coef);
}

// ---------------- elementwise ReLU ----------------
__global__ __launch_bounds__(256) void k_relu(const float* __restrict__ A,
                                              float* __restrict__ O, long n4) {
    long t = (long)blockIdx.x * 256 + threadIdx.x;
    if (t >= n4) return;
    float4 v = ((const float4*)A)[t];
    v.x = fmaxf(v.x, 0.0f);
    v.y = fmaxf(v.y, 0.0f);
    v.z = fmaxf(v.z, 0.0f);
    v.w = fmaxf(v.w, 0.0f);
    ((float4*)O)[t] = v;
}

extern "C" void kernel_launch(void* const* d_in, const int* in_sizes, int n_in,
                              void* d_out, int out_size, void* d_ws, size_t ws_size,
                              hipStream_t stream) {
    const float* x  = (const float*)d_in[0];
    const int*   ei = (const int*)d_in[1];
    const float* W1 = (const float*)d_in[2];
    const float* b1 = (const float*)d_in[3];
    const float* W2 = (const float*)d_in[4];
    const float* b2 = (const float*)d_in[5];
    float* out = (float*)d_out;

    const int n = in_sizes[0] / D;
    const int e = in_sizes[1] / 2;
    const int* row = ei;
    const int* col = ei + e;

    // workspace: dinv[n] | hbuf[n*D] | aggbuf[n*D]
    float* dinv = (float*)d_ws;
    size_t off = (((size_t)n * 4) + 255) & ~(size_t)255;
    float* hbuf = (float*)((char*)d_ws + off);
    float* aggbuf = hbuf + (size_t)n * D;

    const int B = 256;
    dim3 blk(B);
    int gN   = (n + B - 1) / B;
    int gE   = (e + B - 1) / B;
    int gMM  = (n + 127) / 128;
    long tSelf = (long)n * 32;
    int gSelf = (int)((tSelf + B - 1) / B);
    long tScat = (long)e * 32;
    int gScat = (int)((tScat + B - 1) / B);
    long n4 = (long)n * D / 4;
    int gRelu = (int)((n4 + B - 1) / B);

    // normalization (shared by both layers)
    k_deg_init<<<gN, blk, 0, stream>>>(dinv, n);
    k_deg_count<<<gE, blk, 0, stream>>>(col, dinv, e);
    k_rsqrt<<<gN, blk, 0, stream>>>(dinv, n);

    // layer 1
    k_gemm<<<gMM, blk, 0, stream>>>(x, W1, hbuf, n);
    k_self<<<gSelf, blk, 0, stream>>>(hbuf, dinv, b1, aggbuf, n);
    k_scatter<<<gScat, blk, 0, stream>>>(hbuf, row, col, dinv, aggbuf, e);
    k_relu<<<gRelu, blk, 0, stream>>>(aggbuf, hbuf, n4);

    // layer 2 (GEMM output reuses aggbuf; aggregation goes straight to d_out)
    k_gemm<<<gMM, blk, 0, stream>>>(hbuf, W2, aggbuf, n);
    k_self<<<gSelf, blk, 0, stream>>>(aggbuf, dinv, b2, out, n);
    k_scatter<<<gScat, blk, 0, stream>>>(aggbuf, row, col, dinv, out, e);
}